// QuConv2D_MQ_7146825580684
// MI455X (gfx1250) — compile-verified
//
#include <hip/hip_runtime.h>
#include <cstdint>

// Problem constants (from reference)
#define Bn      32
#define Cn      3
#define Hn      256
#define Wdn     256
#define OHn     254
#define OWn     254
#define NQn     9
// Tiling
#define TW      64          // output tile width
#define THT     16          // output tile height
#define LTW     (TW + 2)    // 66: input tile width (halo)
#define LTH     (THT + 2)   // 18: input tile height (halo)
#define TILES_X ((OWn + TW - 1) / TW)    // 4
#define TILES_Y ((OHn + THT - 1) / THT)  // 16

// float(1/(2*pi)) — v_cos_f32 takes input in revolutions (period 1.0)
#define INV2PI 0.15915493667125702f

__global__ __launch_bounds__(256) void quconv2d_mq_kernel(
    const float* __restrict__ x,   // [B, C, H, W]
    const float* __restrict__ Wq,  // [C, NQ]
    float* __restrict__ out)       // [B, C, OH, OW, NQ] flat
{
    __shared__ float tile[LTH * LTW];

    const int t = threadIdx.x;
    unsigned bid = blockIdx.x;
    const int tx = bid % TILES_X; bid /= TILES_X;
    const int ty = bid % TILES_Y; bid /= TILES_Y;
    const int bc = (int)bid;                 // 0 .. B*C-1 (plane index)
    const int c  = bc % Cn;

    const int x0 = tx * TW;
    const int y0 = ty * THT;

    const float* plane = x + (size_t)bc * (Hn * Wdn);

    // ---------------- Async global -> LDS tile staging (CDNA5) ----------------
    // GVS addressing: mem = SGPR64(plane) + VGPR_i32(byte offset). Destination
    // VGPR holds the workgroup-relative LDS byte address (low 32 bits of the
    // generic pointer to __shared__). Edge tiles clamp the source coordinate
    // (duplicated reads; those lanes' outputs are never stored).
    for (int idx = t; idx < LTH * LTW; idx += 256) {
        int ly = idx / LTW;
        int lx = idx - ly * LTW;
        int gy = y0 + ly; if (gy > Hn  - 1) gy = Hn  - 1;
        int gx = x0 + lx; if (gx > Wdn - 1) gx = Wdn - 1;
        unsigned goff = (unsigned)((gy * Wdn + gx) * 4);
        unsigned loff = (unsigned)(uintptr_t)(&tile[idx]);
        asm volatile("global_load_async_to_lds_b32 %0, %1, %2"
                     :: "v"(loff), "v"(goff), "s"(plane)
                     : "memory");
    }
    // Each wave waits for its own async transfers, then the workgroup barrier
    // makes all waves' LDS writes visible to everyone.
    asm volatile("s_wait_asynccnt 0" ::: "memory");
    __syncthreads();

    // ---------------- Compute phase ----------------
    // Per-channel bias, pre-scaled by 1/(2*pi) so each output is one v_fma +
    // one v_cos_f32. c is uniform per block -> scalar loads.
    float wk[NQn];
#pragma unroll
    for (int q = 0; q < NQn; ++q)
        wk[q] = Wq[c * NQn + q] * INV2PI;

    const int px  = t & (TW - 1);     // 0..63  (lane-adjacent => coalesced stores)
    const int py0 = (t >> 6) * 4;     // 0,4,8,12 : 1x4 vertical strip per thread

    const int ox = x0 + px;
    if (ox < OWn) {
        // 6 LDS rows cover the four 3x3 windows of this strip; row sum-of-squares
        // is shared by up to 3 pixels.
        float a[6][3];
        float rsq[6];
#pragma unroll
        for (int r = 0; r < 6; ++r) {
            const float* rp = &tile[(py0 + r) * LTW + px];
            float v0 = rp[0], v1 = rp[1], v2 = rp[2];
            a[r][0] = v0; a[r][1] = v1; a[r][2] = v2;
            rsq[r] = fmaf(v0, v0, fmaf(v1, v1, v2 * v2));
        }

        const size_t obase = (size_t)bc * (OHn * (size_t)OWn) * NQn;
#pragma unroll
        for (int j = 0; j < 4; ++j) {
            const int oy = y0 + py0 + j;
            if (oy < OHn) {
                float s    = rsq[j] + rsq[j + 1] + rsq[j + 2];
                float n    = sqrtf(s);
                float invn = 1.0f / fmaxf(n, 1e-12f);   // matches max(nrm, EPS)
                float ik   = invn * INV2PI;             // fold 1/(2*pi)
                float* op  = out + obase + ((size_t)oy * OWn + ox) * NQn;
#pragma unroll
                for (int dy = 0; dy < 3; ++dy) {
#pragma unroll
                    for (int dx = 0; dx < 3; ++dx) {
                        // cos(p*invn + W) == v_cos( fma(p, invn/2pi, W/2pi) )
                        float u = fmaf(a[j + dy][dx], ik, wk[dy * 3 + dx]);
                        op[dy * 3 + dx] = __builtin_amdgcn_cosf(u);
                    }
                }
            }
        }
    }
}

extern "C" void kernel_launch(void* const* d_in, const int* in_sizes, int n_in,
                              void* d_out, int out_size, void* d_ws, size_t ws_size,
                              hipStream_t stream) {
    const float* x  = (const float*)d_in[0];   // [32,3,256,256] fp32
    const float* Wq = (const float*)d_in[1];   // [3,9] fp32
    float* out      = (float*)d_out;           // 32*27*254*254 fp32

    const int grid = TILES_X * TILES_Y * Bn * Cn;  // 6144 blocks
    quconv2d_mq_kernel<<<grid, 256, 0, stream>>>(x, Wq, out);
}